// AdaGATConv_35098472743569
// MI455X (gfx1250) — compile-verified
//
#include <hip/hip_runtime.h>
#include <math.h>

// GATConv on MI455X (gfx1250, wave32).
// h = x@W via V_WMMA_F32_16X16X4_F32 (exact fp32), then edge softmax +
// atomic scatter (bandwidth-bound part, ~2GB traffic -> ~100us floor @23.3TB/s).

typedef __attribute__((ext_vector_type(2))) float v2f;
typedef __attribute__((ext_vector_type(8))) float v8f;

#define IN_CH 128   // input channels
#define HC    128   // H*C
#define NHEAD 2
#define CCH   64    // channels per head

// ---- monotone float <-> uint key for atomicMax on floats ------------------
__device__ __forceinline__ unsigned flipf(float f) {
  unsigned u = __float_as_uint(f);
  return (u & 0x80000000u) ? ~u : (u | 0x80000000u);
}
__device__ __forceinline__ float unflipf(unsigned u) {
  return (u & 0x80000000u) ? __uint_as_float(u ^ 0x80000000u)
                           : __uint_as_float(~u);
}

// ---- 1) init workspace (must run every call; ws not re-poisoned) ----------
__global__ void gat_init(float* __restrict__ num, float* __restrict__ esum,
                         unsigned* __restrict__ emaxu, int N) {
  int tid = blockIdx.x * blockDim.x + threadIdx.x;
  if (tid < N * HC)    num[tid] = 0.0f;
  if (tid < N * NHEAD) { esum[tid] = 0.0f; emaxu[tid] = 0u; }
}

// ---- 2) h = x @ W  (WMMA f32 16x16x4) -------------------------------------
// Block = 256 threads = 8 waves. Block handles a 16-row strip of x; wave w
// owns output columns [16w, 16w+16). K=128 -> 32 WMMA ops per wave.
__global__ __launch_bounds__(256) void gat_gemm(const float* __restrict__ x,
                                                const float* __restrict__ Wm,
                                                float* __restrict__ h, int N) {
  __shared__ float xs[16][132];  // pad 132: stride%64==4 -> conflict-free cols
  const int rowBase = blockIdx.x * 16;
  const int tid = threadIdx.x;

  // cooperative load of the 16x128 x tile (coalesced)
  #pragma unroll
  for (int i = 0; i < 8; ++i) {
    int idx = tid + i * 256;            // 0..2047
    int r = idx >> 7, c = idx & 127;
    int gr = rowBase + r;
    xs[r][c] = (gr < N) ? x[(size_t)gr * IN_CH + c] : 0.0f;
  }
  __syncthreads();

  const int wave = tid >> 5;
  const int lane = tid & 31;
  const int half = lane >> 4;     // 0: K=0,1 ; 1: K=2,3 (A & B 32-bit layout)
  const int mrow = lane & 15;     // A row / B column within the tile
  const int colBase = wave * 16;

  v8f acc = {};
  #pragma unroll
  for (int k0 = 0; k0 < IN_CH; k0 += 4) {
    const int k = k0 + 2 * half;
    v2f a, b;
    // A 16x4: lanes 0-15 M=lane hold K=k0,k0+1 ; lanes 16-31 hold K=k0+2,k0+3
    a.x = xs[mrow][k];
    a.y = xs[mrow][k + 1];
    // B 4x16: row K striped across lanes (v0: K=k0|k0+2, v1: K=k0+1|k0+3)
    b.x = Wm[(size_t)k * HC + colBase + mrow];
    b.y = Wm[(size_t)(k + 1) * HC + colBase + mrow];
    acc = __builtin_amdgcn_wmma_f32_16x16x4_f32(
        /*neg_a=*/false, a, /*neg_b=*/false, b,
        /*c_mod=*/(short)0, acc, /*reuse_a=*/false, /*reuse_b=*/false);
  }

  // C/D layout: VGPR i -> M = i (lanes 0-15) / i+8 (lanes 16-31), N = lane&15
  #pragma unroll
  for (int i = 0; i < 8; ++i) {
    int gr = rowBase + i + 8 * half;
    if (gr < N) h[(size_t)gr * HC + colBase + mrow] = acc[i];
  }
}

// ---- 3) per-node attention logits ----------------------------------------
__global__ void gat_logits(const float* __restrict__ h,
                           const float* __restrict__ att_src,
                           const float* __restrict__ att_dst,
                           float* __restrict__ a_s, float* __restrict__ a_d,
                           int N) {
  int tid = blockIdx.x * blockDim.x + threadIdx.x;
  if (tid >= N * NHEAD) return;
  int n = tid >> 1, hd = tid & 1;
  const float* hp = h + (size_t)n * HC + hd * CCH;
  const float* ps = att_src + hd * CCH;
  const float* pd = att_dst + hd * CCH;
  float ss = 0.0f, sd = 0.0f;
  #pragma unroll 8
  for (int c = 0; c < CCH; ++c) {
    float v = hp[c];
    ss += v * ps[c];
    sd += v * pd[c];
  }
  a_s[tid] = ss;
  a_d[tid] = sd;
}

// ---- 4) segment max of leaky-relu scores over dst -------------------------
__global__ void gat_edge_max(const int* __restrict__ ei,
                             const float* __restrict__ a_s,
                             const float* __restrict__ a_d,
                             unsigned* __restrict__ emaxu, int E, int N) {
  int tid = blockIdx.x * blockDim.x + threadIdx.x;
  int T = E + N;
  if (tid >= T) return;
  int src = (tid < E) ? ei[tid] : (tid - E);       // self loops appended
  int dst = (tid < E) ? ei[E + tid] : (tid - E);
  #pragma unroll
  for (int hd = 0; hd < NHEAD; ++hd) {
    float s = a_s[src * NHEAD + hd] + a_d[dst * NHEAD + hd];
    s = (s > 0.0f) ? s : 0.2f * s;                 // leaky relu
    atomicMax(&emaxu[dst * NHEAD + hd], flipf(s));
  }
}

// ---- 5) exp weights: esum + unnormalized numerator scatter ----------------
// One wave per edge; lane handles 4 consecutive channels (lane>>4 == head).
__global__ __launch_bounds__(256) void gat_edge_acc(
    const int* __restrict__ ei, const float* __restrict__ a_s,
    const float* __restrict__ a_d, const unsigned* __restrict__ emaxu,
    const float* __restrict__ h, float* __restrict__ esum,
    float* __restrict__ num, int E, int N) {
  int gid = blockIdx.x * 8 + (threadIdx.x >> 5);   // wave id (uniform/wave)
  int T = E + N;
  if (gid >= T) return;
  int lane = threadIdx.x & 31;
  int src = (gid < E) ? ei[gid] : (gid - E);
  int dst = (gid < E) ? ei[E + gid] : (gid - E);
  int hd = lane >> 4;

  float s = a_s[src * NHEAD + hd] + a_d[dst * NHEAD + hd];
  s = (s > 0.0f) ? s : 0.2f * s;
  float m = unflipf(emaxu[dst * NHEAD + hd]);
  float w = __expf(s - m);
  if ((lane & 15) == 0) atomicAdd(&esum[dst * NHEAD + hd], w);

  const float4 hv = *(const float4*)(h + (size_t)src * HC + lane * 4);
  float* np = num + (size_t)dst * HC + lane * 4;
  atomicAdd(np + 0, w * hv.x);
  atomicAdd(np + 1, w * hv.y);
  atomicAdd(np + 2, w * hv.z);
  atomicAdd(np + 3, w * hv.w);
}

// ---- 6) normalize, mean over heads, add bias ------------------------------
__global__ void gat_final(const float* __restrict__ num,
                          const float* __restrict__ esum,
                          const float* __restrict__ bias,
                          float* __restrict__ out, int N) {
  int tid = blockIdx.x * blockDim.x + threadIdx.x;
  if (tid >= N * CCH) return;
  int n = tid >> 6, c = tid & 63;
  float r0 = num[(size_t)n * HC + c]       / (esum[n * NHEAD + 0] + 1e-16f);
  float r1 = num[(size_t)n * HC + CCH + c] / (esum[n * NHEAD + 1] + 1e-16f);
  out[tid] = 0.5f * (r0 + r1) + bias[c];
}

extern "C" void kernel_launch(void* const* d_in, const int* in_sizes, int n_in,
                              void* d_out, int out_size, void* d_ws,
                              size_t ws_size, hipStream_t stream) {
  const float* x       = (const float*)d_in[0];  // [N, 128]
  const float* Wm      = (const float*)d_in[1];  // [128, 128]
  const float* att_src = (const float*)d_in[2];  // [2, 64]
  const float* att_dst = (const float*)d_in[3];  // [2, 64]
  const float* bias    = (const float*)d_in[4];  // [64]
  const int*   ei      = (const int*)d_in[5];    // [2, E]

  const int N = in_sizes[0] / IN_CH;
  const int E = in_sizes[5] / 2;
  const int T = E + N;

  // workspace layout (~53 MB)
  float* h      = (float*)d_ws;                       // N*128
  float* num    = h + (size_t)N * HC;                 // N*128
  float* a_s    = num + (size_t)N * HC;               // N*2
  float* a_d    = a_s + (size_t)N * NHEAD;            // N*2
  float* esum   = a_d + (size_t)N * NHEAD;            // N*2
  unsigned* emx = (unsigned*)(esum + (size_t)N * NHEAD); // N*2
  float* out    = (float*)d_out;

  gat_init    <<<(N * HC + 255) / 256, 256, 0, stream>>>(num, esum, emx, N);
  gat_gemm    <<<(N + 15) / 16,        256, 0, stream>>>(x, Wm, h, N);
  gat_logits  <<<(N * NHEAD + 255) / 256, 256, 0, stream>>>(h, att_src, att_dst,
                                                            a_s, a_d, N);
  gat_edge_max<<<(T + 255) / 256,      256, 0, stream>>>(ei, a_s, a_d, emx, E, N);
  gat_edge_acc<<<(T + 7) / 8,          256, 0, stream>>>(ei, a_s, a_d, emx, h,
                                                         esum, num, E, N);
  gat_final   <<<(N * CCH + 255) / 256, 256, 0, stream>>>(num, esum, bias, out, N);
}